// PLGRL_88725434400997
// MI455X (gfx1250) — compile-verified
//
#include <hip/hip_runtime.h>
#include <hip/hip_bf16.h>
#include <math.h>

typedef float v2f __attribute__((ext_vector_type(2)));
typedef float v8f __attribute__((ext_vector_type(8)));

#define POOL  20
#define NNODE 400
#define CFEAT 512
#define NCLS  21

#define GF_BIAS 1
#define GF_RELU 2
#define GF_CT   4   // store C transposed: C[ncol*ldc + row]

// ---------------------------------------------------------------------------
// 1) Adaptive avg pool 400x400 -> 20x20, writing feat[n][c] with the
//    reference's transposed node order: n = wi*20 + hi, feat[n][c]=pool[c][hi][wi]
//    One block per (c, hi): 400 threads each own one input column, coalesced.
// ---------------------------------------------------------------------------
__global__ void pool_kernel(const float* __restrict__ x, float* __restrict__ feat) {
  int c  = blockIdx.x / POOL;
  int ph = blockIdx.x % POOL;            // hi
  __shared__ float colacc[400];
  int t = threadIdx.x;                   // column 0..399
  const float* base = x + ((size_t)c * 400 + (size_t)ph * POOL) * 400;
  float acc = 0.f;
#pragma unroll
  for (int r = 0; r < POOL; ++r) acc += base[r * 400 + t];
  colacc[t] = acc;
  __syncthreads();
  if (t < POOL) {                        // t = wi (column block)
    float s = 0.f;
#pragma unroll
    for (int k = 0; k < POOL; ++k) s += colacc[t * POOL + k];
    feat[(size_t)(t * POOL + ph) * CFEAT + c] = s * (1.0f / 400.0f);
  }
}

// ---------------------------------------------------------------------------
// 2) Row L2 norms (clamped at 1e-8)
// ---------------------------------------------------------------------------
__global__ void norm_kernel(const float* __restrict__ feat, float* __restrict__ norms) {
  __shared__ float red[256];
  int n = blockIdx.x, t = threadIdx.x;
  float s = 0.f;
  for (int c = t; c < CFEAT; c += 256) { float v = feat[(size_t)n * CFEAT + c]; s += v * v; }
  red[t] = s; __syncthreads();
  for (int off = 128; off > 0; off >>= 1) { if (t < off) red[t] += red[t + off]; __syncthreads(); }
  if (t == 0) norms[n] = fmaxf(sqrtf(red[0]), 1e-8f);
}

// ---------------------------------------------------------------------------
// 3) Rasterize gt boxes into isbox / label grids (later boxes overwrite)
// ---------------------------------------------------------------------------
__global__ void box_kernel(const float* __restrict__ im_info, const float* __restrict__ gt,
                           const int* __restrict__ nbox, float* __restrict__ isbv,
                           int* __restrict__ labels) {
  int n = threadIdx.x;
  if (n >= NNODE) return;
  float yy = (float)(n / POOL), xx = (float)(n % POOL);
  float H = im_info[0], W = im_info[1];
  int nb = nbox[0];
  float isb = 0.f, lab = 0.f;
  for (int j = 0; j < nb; ++j) {
    const float* g = gt + j * 5;
    float x1i = floorf(g[0] * (float)POOL / W);
    float y1i = floorf(g[1] * (float)POOL / H);
    float x2i = floorf(g[2] * (float)POOL / W);
    float y2i = floorf(g[3] * (float)POOL / H);
    if (yy >= y1i && yy < y2i && xx >= x1i && xx < x2i) { isb = 1.f; lab = g[4]; }
  }
  isbv[n] = isb;
  labels[n] = (int)lab;
}

// ---------------------------------------------------------------------------
// 4) Out-of-place transpose: out[C,R] = in[R,C]^T  (tiny matrices, one pass)
// ---------------------------------------------------------------------------
__global__ void transpose_kernel(const float* __restrict__ in, float* __restrict__ out,
                                 int R, int C) {
  int idx = blockIdx.x * blockDim.x + threadIdx.x;
  if (idx >= R * C) return;
  int r = idx / C, c = idx % C;
  out[(size_t)c * R + r] = in[idx];
}

// ---------------------------------------------------------------------------
// 5) fp32 WMMA GEMM, B always transposed: C[M,N] = A[M,K] @ Bt[N,K]^T
//    (+bias per column, optional relu, optional transposed store).
//    One 16x16 tile per wave via V_WMMA_F32_16X16X4_F32; K % 4 == 0.
//    Ragged edges handled by clamped-address loads (branch-free; garbage only
//    reaches rows/cols that the guarded store never writes).
// ---------------------------------------------------------------------------
__global__ __launch_bounds__(256)
void gemm_wmma_f32(const float* __restrict__ A, const float* __restrict__ Bt,
                   const float* __restrict__ bias, float* __restrict__ C,
                   int M, int N, int K, int lda, int ldb, int ldc, int flags) {
  int wave = threadIdx.x >> 5;
  int lane = threadIdx.x & 31;
  int tilesN = (N + 15) >> 4;
  int tilesM = (M + 15) >> 4;
  int tile = blockIdx.x * (blockDim.x >> 5) + wave;
  if (tile >= tilesM * tilesN) return;       // wave-uniform: EXEC stays all-1s
  int tm = (tile / tilesN) << 4;
  int tn = (tile % tilesN) << 4;
  int l16  = lane & 15;
  int half = lane >> 4;                      // 0 or 1
  int mrow = tm + l16;
  int ncol = tn + l16;
  int kh   = half * 2;
  bool nok = ncol < N;
  int mrow_c = (mrow < M) ? mrow : (M - 1);  // clamped coords -> no exec masks
  int ncol_c = nok ? ncol : (N - 1);

  const float* aptr = A  + (size_t)mrow_c * lda + kh;  // even offsets: 8B aligned
  const float* bptr = Bt + (size_t)ncol_c * ldb + kh;
  v8f acc = {0.f, 0.f, 0.f, 0.f, 0.f, 0.f, 0.f, 0.f};

#pragma unroll 4
  for (int k0 = 0; k0 < K; k0 += 4) {
    v2f a = *(const v2f*)aptr;
    v2f b = *(const v2f*)bptr;
    aptr += 4; bptr += 4;
    acc = __builtin_amdgcn_wmma_f32_16x16x4_f32(
        false, a, false, b, (short)0, acc, false, false);
  }

  float bv = 0.f;
  if ((flags & GF_BIAS) && nok) bv = bias[ncol];
  if (flags & GF_CT) {
    float* cp = C + (size_t)ncol * ldc + tm + half * 8;  // 8 consecutive floats
#pragma unroll
    for (int r = 0; r < 8; ++r) {
      int row = tm + r + half * 8;
      if (row < M && nok) {
        float v = acc[r] + bv;
        if (flags & GF_RELU) v = fmaxf(v, 0.0f);
        cp[r] = v;
      }
    }
  } else {
#pragma unroll
    for (int r = 0; r < 8; ++r) {
      int row = tm + r + half * 8;
      if (row < M && nok) {
        float v = acc[r] + bv;
        if (flags & GF_RELU) v = fmaxf(v, 0.0f);
        C[(size_t)row * ldc + ncol] = v;
      }
    }
  }
}

// ---------------------------------------------------------------------------
// 6) Similarity epilogue (in place on S = feat@feat^T):
//    |cos| - eye, threshold 0.5, * dis * isb_i * isb_j, + eye
// ---------------------------------------------------------------------------
__global__ void simi_kernel(float* __restrict__ S, const float* __restrict__ norms,
                            const float* __restrict__ isbv) {
  int idx = blockIdx.x * blockDim.x + threadIdx.x;
  if (idx >= NNODE * NNODE) return;
  int i = idx / NNODE, j = idx % NNODE;
  float v = fabsf(S[idx]) / (norms[i] * norms[j]);
  if (i == j) v -= 1.0f;
  if (v < 0.5f) v = 0.0f;
  float dy = (float)(i / POOL) - (float)(j / POOL);
  float dx = (float)(i % POOL) - (float)(j % POOL);
  float dis = (dy * dy + dx * dx < 50.0f) ? 1.0f : 0.0f;
  v *= dis * isbv[i] * isbv[j];
  if (i == j) v += 1.0f;
  S[idx] = v;
}

// 7) d_i = rowsum^-1/2
__global__ void rowsum_kernel(const float* __restrict__ S, float* __restrict__ dinv) {
  __shared__ float red[256];
  int i = blockIdx.x, t = threadIdx.x;
  float s = 0.f;
  for (int j = t; j < NNODE; j += 256) s += S[(size_t)i * NNODE + j];
  red[t] = s; __syncthreads();
  for (int off = 128; off > 0; off >>= 1) { if (t < off) red[t] += red[t + off]; __syncthreads(); }
  if (t == 0) dinv[i] = rsqrtf(red[0]);
}

// 8) A = D^-1/2 S D^-1/2
__global__ void scaleA_kernel(const float* __restrict__ S, const float* __restrict__ dinv,
                              float* __restrict__ Amat) {
  int idx = blockIdx.x * blockDim.x + threadIdx.x;
  if (idx >= NNODE * NNODE) return;
  int i = idx / NNODE, j = idx % NNODE;
  Amat[idx] = dinv[i] * S[idx] * dinv[j];
}

// 9) mean NLL over log_softmax(logits)
__global__ void loss_kernel(const float* __restrict__ logits, const int* __restrict__ labels,
                            float* __restrict__ out) {
  __shared__ float red[512];
  int t = threadIdx.x;
  float l = 0.f;
  if (t < NNODE) {
    const float* row = logits + (size_t)t * NCLS;
    float m = row[0];
    for (int k = 1; k < NCLS; ++k) m = fmaxf(m, row[k]);
    float s = 0.f;
    for (int k = 0; k < NCLS; ++k) s += __expf(row[k] - m);
    float lse = m + __logf(s);
    l = lse - row[labels[t]];
  }
  red[t] = l; __syncthreads();
  for (int off = 256; off > 0; off >>= 1) { if (t < off) red[t] += red[t + off]; __syncthreads(); }
  if (t == 0) out[0] = red[0] * (1.0f / (float)NNODE);
}

// ---------------------------------------------------------------------------
extern "C" void kernel_launch(void* const* d_in, const int* in_sizes, int n_in,
                              void* d_out, int out_size, void* d_ws, size_t ws_size,
                              hipStream_t stream) {
  const float* x       = (const float*)d_in[0];
  const float* im_info = (const float*)d_in[1];
  const float* gt      = (const float*)d_in[2];
  const int*   nbox    = (const int*)d_in[3];
  const float* W1  = (const float*)d_in[4];
  const float* b1  = (const float*)d_in[5];
  const float* W2  = (const float*)d_in[6];
  const float* b2  = (const float*)d_in[7];
  const float* fw1 = (const float*)d_in[8];
  const float* fb1 = (const float*)d_in[9];
  const float* fw2 = (const float*)d_in[10];
  const float* fb2 = (const float*)d_in[11];
  const float* fw3 = (const float*)d_in[12];
  const float* fb3 = (const float*)d_in[13];

  float* ws = (float*)d_ws;
  size_t off = 0;
  float* feat   = ws + off; off += (size_t)NNODE * CFEAT;   // [400,512]
  float* S      = ws + off; off += (size_t)NNODE * NNODE;   // [400,400]
  float* Amat   = ws + off; off += (size_t)NNODE * NNODE;   // [400,400]
  float* norms  = ws + off; off += NNODE;
  float* isbv   = ws + off; off += NNODE;
  float* dinv   = ws + off; off += NNODE;
  int*   labels = (int*)(ws + off); off += NNODE;
  float* W1t    = ws + off; off += (size_t)256 * 512;       // W1^T  [256,512]
  float* W2t    = ws + off; off += (size_t)128 * 256;       // W2^T  [128,256]
  float* fw1t   = ws + off; off += (size_t)256 * 128;       // fw1^T [256,128]
  float* fw2t   = ws + off; off += (size_t)512 * 256;       // fw2^T [512,256]
  float* fw3t   = ws + off; off += (size_t)NCLS * 512;      // fw3^T [21,512]
  float* XW1t   = ws + off; off += (size_t)256 * NNODE;     // (feat@W1)^T [256,400]
  float* h1     = ws + off; off += (size_t)NNODE * 256;     // [400,256]
  float* HW2t   = ws + off; off += (size_t)128 * NNODE;     // (h1@W2)^T   [128,400]
  float* h2     = ws + off; off += (size_t)NNODE * 128;     // [400,128]
  float* z1     = ws + off; off += (size_t)NNODE * 256;     // [400,256]
  float* z2     = ws + off; off += (size_t)NNODE * 512;     // [400,512]
  float* logits = ws + off; off += (size_t)NNODE * NCLS;    // [400,21]

  auto gblocks = [](int M, int N) {
    int tiles = ((M + 15) / 16) * ((N + 15) / 16);
    return (tiles + 7) / 8;            // 8 waves / 256-thread block
  };
  auto tblocks = [](int n) { return (n + 255) / 256; };

  // Stage 1: pool (bandwidth-dominant, single coalesced pass over 328 MB)
  pool_kernel<<<CFEAT * POOL, 400, 0, stream>>>(x, feat);
  norm_kernel<<<NNODE, 256, 0, stream>>>(feat, norms);
  box_kernel<<<1, 400, 0, stream>>>(im_info, gt, nbox, isbv, labels);

  // Stage 1b: transpose weights once so every GEMM uses the fast A·B^T loop
  transpose_kernel<<<tblocks(512 * 256), 256, 0, stream>>>(W1,  W1t,  512, 256);
  transpose_kernel<<<tblocks(256 * 128), 256, 0, stream>>>(W2,  W2t,  256, 128);
  transpose_kernel<<<tblocks(128 * 256), 256, 0, stream>>>(fw1, fw1t, 128, 256);
  transpose_kernel<<<tblocks(256 * 512), 256, 0, stream>>>(fw2, fw2t, 256, 512);
  transpose_kernel<<<tblocks(512 * NCLS), 256, 0, stream>>>(fw3, fw3t, 512, NCLS);

  // Stage 2: S = feat @ feat^T  (400x400x512)
  gemm_wmma_f32<<<gblocks(NNODE, NNODE), 256, 0, stream>>>(
      feat, feat, nullptr, S, NNODE, NNODE, CFEAT, CFEAT, CFEAT, NNODE, 0);
  simi_kernel<<<tblocks(NNODE * NNODE), 256, 0, stream>>>(S, norms, isbv);
  rowsum_kernel<<<NNODE, 256, 0, stream>>>(S, dinv);
  scaleA_kernel<<<tblocks(NNODE * NNODE), 256, 0, stream>>>(S, dinv, Amat);

  // Stage 3: GCN head. Producers of A@(.) operands store transposed (GF_CT)
  // so consumers read them as [N,K] B^T operands with b64 loads.
  gemm_wmma_f32<<<gblocks(NNODE, 256), 256, 0, stream>>>(
      feat, W1t, nullptr, XW1t, NNODE, 256, 512, 512, 512, NNODE, GF_CT);
  gemm_wmma_f32<<<gblocks(NNODE, 256), 256, 0, stream>>>(
      Amat, XW1t, b1, h1, NNODE, 256, NNODE, NNODE, NNODE, 256, GF_BIAS | GF_RELU);
  gemm_wmma_f32<<<gblocks(NNODE, 128), 256, 0, stream>>>(
      h1, W2t, nullptr, HW2t, NNODE, 128, 256, 256, 256, NNODE, GF_CT);
  gemm_wmma_f32<<<gblocks(NNODE, 128), 256, 0, stream>>>(
      Amat, HW2t, b2, h2, NNODE, 128, NNODE, NNODE, NNODE, 128, GF_BIAS);

  // Stage 4: MLP head
  gemm_wmma_f32<<<gblocks(NNODE, 256), 256, 0, stream>>>(
      h2, fw1t, fb1, z1, NNODE, 256, 128, 128, 128, 256, GF_BIAS | GF_RELU);
  gemm_wmma_f32<<<gblocks(NNODE, 512), 256, 0, stream>>>(
      z1, fw2t, fb2, z2, NNODE, 512, 256, 256, 256, 512, GF_BIAS | GF_RELU);
  gemm_wmma_f32<<<gblocks(NNODE, NCLS), 256, 0, stream>>>(
      z2, fw3t, fb3, logits, NNODE, NCLS, 512, 512, 512, NCLS, GF_BIAS);

  // Stage 5: loss
  loss_kernel<<<1, 512, 0, stream>>>(logits, labels, (float*)d_out);
}